// LineSpectralPairsToLinearPredictiveCoefficients_55009941127437
// MI455X (gfx1250) — compile-verified
//
#include <hip/hip_runtime.h>

// LSP -> LPC, M = 24.
// Row layout: [K, w1..w24] (25 f32). Output row: [K, a0..a23] (25 f32).
// P(x) = prod_k (x^2 - 2cos(w[2+2k]) x + 1)   (12 quadratics)
// Q(x) = prod_k (x^2 - 2cos(w[1+2k]) x + 1)   (12 quadratics)
// a_i  = 0.5*((P[i+1]-P[i]) + (Q[i+1]+Q[i]))  (coeffs in descending powers)
// Both polys are palindromic -> carry only half (13) coefficients.

#define NC 25
#define NQ 12
#define ROWS_PER_BLOCK 256
#define TILE_DW (ROWS_PER_BLOCK * NC)   // 6400 dwords = 25600 bytes per block
#define TILE_XFERS (TILE_DW / 4)        // 1600 b128 transfers = 6*256 + 64

// Async global->LDS / LDS->global, b128 per lane. INST_OFFSET is added to both
// the LDS address (VDST) and the global address, so one base VGPR (pair) serves
// all unrolled transfers; only the 24-bit immediate changes.
#define ASYNC_LOAD_B128(ldsv, gav, imm)                                        \
    asm volatile("global_load_async_to_lds_b128 %0, %1, off offset:" #imm      \
                 :: "v"(ldsv), "v"(gav) : "memory")
#define ASYNC_STORE_B128(gav, ldsv, imm)                                       \
    asm volatile("global_store_async_from_lds_b128 %0, %1, off offset:" #imm   \
                 " th:TH_STORE_NT"                                             \
                 :: "v"(gav), "v"(ldsv) : "memory")

// Build half of the palindromic product of quadratics (x^2 + a_k x + 1).
// On exit h[0..12] = c[0..12] of the degree-24 monic palindromic polynomial.
__device__ __forceinline__ void half_palindromic_poly(const float a[NQ], float h[NQ + 1]) {
    h[0] = 1.0f;
    h[1] = a[0];
#pragma unroll
    for (int k = 1; k < NQ; ++k) {
        const float ak = a[k];
        // new middle coefficient (index k+1): c[k+1] + ak*c[k] + c[k-1], c[k+1]==c[k-1]
        const float nm = __builtin_fmaf(ak, h[k], 2.0f * h[k - 1]);
#pragma unroll
        for (int j = k; j >= 2; --j)
            h[j] = __builtin_fmaf(ak, h[j - 1], h[j] + h[j - 2]);
        h[1] = __builtin_fmaf(ak, h[0], h[1]);
        h[k + 1] = nm;
    }
}

__device__ __forceinline__ void lsp_row(const float v[NC], float o[NC]) {
    float ap[NQ], aq[NQ];
#pragma unroll
    for (int k = 0; k < NQ; ++k) {
        aq[k] = -2.0f * __cosf(v[1 + 2 * k]);   // q roots: w indices 1,3,...,23
        ap[k] = -2.0f * __cosf(v[2 + 2 * k]);   // p roots: w indices 2,4,...,24
    }
    float hp[NQ + 1], hq[NQ + 1];
    half_palindromic_poly(ap, hp);
    half_palindromic_poly(aq, hq);

    o[0] = v[0];                                 // K passes through
#pragma unroll
    for (int i = 0; i < 24; ++i) {
        const int i0 = (i <= 12) ? i : 24 - i;           // P[i]   via palindrome
        const int i1 = (i + 1 <= 12) ? i + 1 : 23 - i;   // P[i+1] via palindrome
        const float pc = hp[i1] - hp[i0];
        const float qc = hq[i1] + hq[i0];
        o[1 + i] = 0.5f * (pc + qc);
    }
}

__global__ void __launch_bounds__(ROWS_PER_BLOCK)
lsp2lpc_tiled(const float* __restrict__ w, float* __restrict__ out) {
    __shared__ __align__(16) float tile[TILE_DW];
    const int tid = threadIdx.x;
    const size_t tile_base = (size_t)blockIdx.x * (size_t)TILE_DW;

    // Per-thread fixed base addresses; unrolled transfers vary only INST_OFFSET.
    const unsigned lds_b = (unsigned)(size_t)(&tile[0]) + (unsigned)tid * 16u;
    const unsigned long long gin = (unsigned long long)(w + tile_base)
                                 + (unsigned long long)tid * 16ull;

    // ---- async global -> LDS staging: 6 full sweeps + 64-transfer tail ----
    ASYNC_LOAD_B128(lds_b, gin, 0);
    ASYNC_LOAD_B128(lds_b, gin, 4096);
    ASYNC_LOAD_B128(lds_b, gin, 8192);
    ASYNC_LOAD_B128(lds_b, gin, 12288);
    ASYNC_LOAD_B128(lds_b, gin, 16384);
    ASYNC_LOAD_B128(lds_b, gin, 20480);
    if (tid < (TILE_XFERS - 6 * ROWS_PER_BLOCK)) {   // tid < 64: waves 0,1 only
        ASYNC_LOAD_B128(lds_b, gin, 24576);
    }
    asm volatile("s_wait_asynccnt 0" ::: "memory");
    __syncthreads();

    // ---- per-lane compute; stride-25 LDS rows hit 32 distinct banks ----
    float v[NC], o[NC];
#pragma unroll
    for (int j = 0; j < NC; ++j) v[j] = tile[tid * NC + j];
    lsp_row(v, o);
#pragma unroll
    for (int j = 0; j < NC; ++j) tile[tid * NC + j] = o[j];
    __syncthreads();

    // ---- async LDS -> global store, non-temporal (write-once stream) ----
    const unsigned long long gout = (unsigned long long)(out + tile_base)
                                  + (unsigned long long)tid * 16ull;
    ASYNC_STORE_B128(gout, lds_b, 0);
    ASYNC_STORE_B128(gout, lds_b, 4096);
    ASYNC_STORE_B128(gout, lds_b, 8192);
    ASYNC_STORE_B128(gout, lds_b, 12288);
    ASYNC_STORE_B128(gout, lds_b, 16384);
    ASYNC_STORE_B128(gout, lds_b, 20480);
    if (tid < (TILE_XFERS - 6 * ROWS_PER_BLOCK)) {
        ASYNC_STORE_B128(gout, lds_b, 24576);
    }
    asm volatile("s_wait_asynccnt 0" ::: "memory");
}

// Generic tail path (plain global loads/stores) for row counts not divisible
// by ROWS_PER_BLOCK. Not used for the harness shape (1,048,576 rows).
__global__ void lsp2lpc_tail(const float* __restrict__ w, float* __restrict__ out,
                             long long row0, long long nrows) {
    const long long r = row0 + (long long)blockIdx.x * blockDim.x + threadIdx.x;
    if (r >= nrows) return;
    const float* src = w + r * NC;
    float v[NC], o[NC];
#pragma unroll
    for (int j = 0; j < NC; ++j) v[j] = src[j];
    lsp_row(v, o);
    float* dst = out + r * NC;
#pragma unroll
    for (int j = 0; j < NC; ++j) dst[j] = o[j];
}

extern "C" void kernel_launch(void* const* d_in, const int* in_sizes, int n_in,
                              void* d_out, int out_size, void* d_ws, size_t ws_size,
                              hipStream_t stream) {
    const float* w = (const float*)d_in[0];
    float* out = (float*)d_out;

    const long long total = (long long)in_sizes[0];   // 16*65536*25
    const long long rows = total / NC;                // 1,048,576
    const long long full_blocks = rows / ROWS_PER_BLOCK;

    if (full_blocks > 0) {
        lsp2lpc_tiled<<<(unsigned)full_blocks, ROWS_PER_BLOCK, 0, stream>>>(w, out);
    }
    const long long rem = rows - full_blocks * ROWS_PER_BLOCK;
    if (rem > 0) {
        const unsigned g = (unsigned)((rem + 255) / 256);
        lsp2lpc_tail<<<g, 256, 0, stream>>>(w, out, full_blocks * ROWS_PER_BLOCK, rows);
    }
}